// ConfigurableNoisyQuantumLayer_66640712565477
// MI455X (gfx1250) — compile-verified
//
#include <hip/hip_runtime.h>
#include <math.h>

typedef float v2f __attribute__((ext_vector_type(2)));
typedef float v8f __attribute__((ext_vector_type(8)));

#define NQ 8
#define DEPTH 6
#define BATCH 32
#define G1 0.0003f
#define G2 0.0065f

// ---------------- complex helpers (superop construction) ----------------
struct c2 { float re, im; };
__device__ __forceinline__ c2 cmul(c2 a, c2 b) {
    return { a.re * b.re - a.im * b.im, a.re * b.im + a.im * b.re };
}
__device__ __forceinline__ c2 cconj(c2 a) { return { a.re, -a.im }; }

__device__ void zero4(c2 S[4][4]) {
    for (int i = 0; i < 4; i++) for (int j = 0; j < 4; j++) S[i][j] = { 0.f, 0.f };
}

// S[(2r+c),(2r'+c')] += K[r][r'] * conj(K[c][c'])
__device__ void kraus_accum(const c2 K[2][2], c2 S[4][4]) {
    for (int r = 0; r < 2; r++)
        for (int c = 0; c < 2; c++)
            for (int rp = 0; rp < 2; rp++)
                for (int cp = 0; cp < 2; cp++) {
                    c2 v = cmul(K[r][rp], cconj(K[c][cp]));
                    S[r * 2 + c][rp * 2 + cp].re += v.re;
                    S[r * 2 + c][rp * 2 + cp].im += v.im;
                }
}

__device__ void m4mul(const c2 A[4][4], const c2 B[4][4], c2 C[4][4]) {
    for (int i = 0; i < 4; i++)
        for (int j = 0; j < 4; j++) {
            c2 acc = { 0.f, 0.f };
            for (int k = 0; k < 4; k++) {
                c2 v = cmul(A[i][k], B[k][j]);
                acc.re += v.re; acc.im += v.im;
            }
            C[i][j] = acc;
        }
}

// composite noise superop: depol(pd) o phase(gp) o amp(ga)  (amp applied first)
__device__ void build_noise(float ga, float gp, float pd, c2 out[4][4]) {
    c2 Sa[4][4], Sp[4][4], Sd[4][4], T[4][4];
    zero4(Sa);
    {
        c2 K0[2][2] = { { {1.f,0.f}, {0.f,0.f} }, { {0.f,0.f}, {sqrtf(1.f - ga),0.f} } };
        kraus_accum(K0, Sa);
        c2 K1[2][2] = { { {0.f,0.f}, {sqrtf(ga),0.f} }, { {0.f,0.f}, {0.f,0.f} } };
        kraus_accum(K1, Sa);
    }
    zero4(Sp);
    {
        c2 K0[2][2] = { { {1.f,0.f}, {0.f,0.f} }, { {0.f,0.f}, {sqrtf(1.f - gp),0.f} } };
        kraus_accum(K0, Sp);
        c2 K1[2][2] = { { {0.f,0.f}, {0.f,0.f} }, { {0.f,0.f}, {sqrtf(gp),0.f} } };
        kraus_accum(K1, Sp);
    }
    zero4(Sd);
    {
        float s0 = sqrtf(1.f - pd), s = sqrtf(pd / 3.f);
        c2 KI[2][2] = { { {s0,0.f}, {0.f,0.f} }, { {0.f,0.f}, {s0,0.f} } };
        kraus_accum(KI, Sd);
        c2 KX[2][2] = { { {0.f,0.f}, {s,0.f} }, { {s,0.f}, {0.f,0.f} } };
        kraus_accum(KX, Sd);
        c2 KY[2][2] = { { {0.f,0.f}, {0.f,-s} }, { {0.f,s}, {0.f,0.f} } };
        kraus_accum(KY, Sd);
        c2 KZ[2][2] = { { {s,0.f}, {0.f,0.f} }, { {0.f,0.f}, {-s,0.f} } };
        kraus_accum(KZ, Sd);
    }
    m4mul(Sp, Sa, T);   // phase o amp
    m4mul(Sd, T, out);  // depol o phase o amp
}

__device__ void super_ry(float t, c2 S[4][4]) {
    zero4(S);
    float c = cosf(0.5f * t), s = sinf(0.5f * t);
    c2 U[2][2] = { { {c,0.f}, {-s,0.f} }, { {s,0.f}, {c,0.f} } };
    kraus_accum(U, S);
}
__device__ void super_rz(float t, c2 S[4][4]) {
    zero4(S);
    float c = cosf(0.5f * t), s = sinf(0.5f * t);
    c2 U[2][2] = { { {c,-s}, {0.f,0.f} }, { {0.f,0.f}, {c,s} } };  // diag(e^{-it/2}, e^{it/2})
    kraus_accum(U, S);
}

__device__ void wire_encode(float theta, c2 out[4][4]) {
    c2 Sr[4][4], N[4][4];
    super_ry(theta, Sr);
    build_noise(G1 * 0.3f, G1 * 0.2f, G1 * 0.5f, N);
    m4mul(N, Sr, out);
}
__device__ void wire_rot(float t_ry, float t_rz, c2 out[4][4]) {
    c2 Sy[4][4], Sz[4][4], T[4][4], N[4][4];
    super_ry(t_ry, Sy);
    super_rz(t_rz, Sz);
    m4mul(Sz, Sy, T);  // rz o ry
    build_noise(G1 * 0.3f, G1 * 0.2f, G1 * 0.5f, N);
    m4mul(N, T, out);  // noise o rz o ry
}

// T16 = kron(Sa, Sb); write row-major re plane [256] + im plane [256]
__device__ void write_kron(const c2 Sa[4][4], const c2 Sb[4][4], float* Tre, float* Tim) {
    for (int mi = 0; mi < 4; mi++)
        for (int mj = 0; mj < 4; mj++)
            for (int ni = 0; ni < 4; ni++)
                for (int nj = 0; nj < 4; nj++) {
                    c2 v = cmul(Sa[mi][ni], Sb[mj][nj]);
                    int M = mi * 4 + mj, N = ni * 4 + nj;
                    Tre[M * 16 + N] = v.re;
                    Tim[M * 16 + N] = v.im;
                }
}

// ---------------- superop build kernel (tiny) ----------------
__global__ __launch_bounds__(256) void build_superops(const float* __restrict__ x,
                                                      const float* __restrict__ w,
                                                      float* enc, float* rot, float* cnotT) {
    int tid = threadIdx.x;
    if (tid < BATCH * 4) {                    // per-sample encode pair superops (real)
        int s = tid >> 2, j = tid & 3;
        c2 Sa[4][4], Sb[4][4];
        wire_encode(x[s * NQ + 2 * j], Sa);
        wire_encode(x[s * NQ + 2 * j + 1], Sb);
        float* T = enc + (s * 4 + j) * 512;
        write_kron(Sa, Sb, T, T + 256);
    } else if (tid < BATCH * 4 + DEPTH * 4) { // per-layer rotation pair superops (complex)
        int idx = tid - BATCH * 4;
        int l = idx >> 2, j = idx & 3;
        c2 Sa[4][4], Sb[4][4];
        wire_rot(w[(l * NQ + 2 * j) * 2 + 0], w[(l * NQ + 2 * j) * 2 + 1], Sa);
        wire_rot(w[(l * NQ + 2 * j + 1) * 2 + 0], w[(l * NQ + 2 * j + 1) * 2 + 1], Sb);
        float* T = rot + (l * 4 + j) * 512;
        write_kron(Sa, Sb, T, T + 256);
    } else if (tid == BATCH * 4 + DEPTH * 4) { // constant CNOT+noise superop (real)
        c2 N2[4][4];
        build_noise(G2 * 0.3f, G2 * 0.2f, G2 * 0.5f, N2);
        // S16 = kron(N2, N2) @ SuperCNOT; SuperCNOT is a column permutation P(N)
        for (int M = 0; M < 16; M++)
            for (int N = 0; N < 16; N++) {
                int ni = N >> 2, nj = N & 3;
                int rj = (nj >> 1) ^ (ni >> 1);   // target row bit ^= control row bit
                int cj = (nj & 1) ^ (ni & 1);     // target col bit ^= control col bit
                int pj = rj * 2 + cj;
                c2 v = cmul(N2[M >> 2][ni], N2[M & 3][pj]);
                cnotT[M * 16 + N] = v.re;
                cnotT[256 + M * 16 + N] = v.im;
            }
    }
}

// ---------------- WMMA superop application ----------------
#define WMMA4(A, B, C) \
    __builtin_amdgcn_wmma_f32_16x16x4_f32(false, (A), false, (B), (short)0, (C), false, false)

// Apply 16x16 superop T (re/im) to the pair axis of wires (i, i+1).
// rho superindex: m = sum_w (2*r_w + c_w) * 4^(7-w); pair-axis stride = 4^(6-i),
// suffix size = 4^(6-i), prefix stride = 4^(8-i). 4096 rest-columns -> 256 tiles of 16.
__device__ void apply_pair_op(float* __restrict__ rre, float* __restrict__ rim,
                              const float* __restrict__ Tre, const float* __restrict__ Tim,
                              int i, bool cplx, int lane, int wave) {
    const int suf_sh = 2 * (6 - i);
    const int m = lane & 15;
    const int kb = (lane < 16) ? 0 : 2;       // A: 16x4 f32 layout (K pair per half-wave)
    const int r0 = (lane < 16) ? 0 : 1;       // B: 4x16 f32 layout (K per VGPR/half-wave)
    const int r1 = (lane < 16) ? 2 : 3;

    v2f aRe[4], aIm[4], aImN[4];
#pragma unroll
    for (int kk = 0; kk < 4; kk++) {
        aRe[kk] = (v2f){ Tre[m * 16 + 4 * kk + kb], Tre[m * 16 + 4 * kk + kb + 1] };
        if (cplx) {
            aIm[kk]  = (v2f){ Tim[m * 16 + 4 * kk + kb], Tim[m * 16 + 4 * kk + kb + 1] };
            aImN[kk] = -aIm[kk];
        }
    }

    for (int t = wave; t < 256; t += 8) {
        const int col  = t * 16 + (lane & 15);
        const int p    = col >> suf_sh;
        const int s    = col & ((1 << suf_sh) - 1);
        const int base = (p << (suf_sh + 4)) + s;

        v2f bRe[4], bIm[4];
#pragma unroll
        for (int kk = 0; kk < 4; kk++) {
            const int e0 = base + ((4 * kk + r0) << suf_sh);
            const int e1 = base + ((4 * kk + r1) << suf_sh);
            bRe[kk] = (v2f){ rre[e0], rre[e1] };
            bIm[kk] = (v2f){ rim[e0], rim[e1] };
        }

        v8f cr = { 0.f, 0.f, 0.f, 0.f, 0.f, 0.f, 0.f, 0.f };
        v8f ci = { 0.f, 0.f, 0.f, 0.f, 0.f, 0.f, 0.f, 0.f };
#pragma unroll
        for (int kk = 0; kk < 4; kk++) cr = WMMA4(aRe[kk], bRe[kk], cr);
#pragma unroll
        for (int kk = 0; kk < 4; kk++) ci = WMMA4(aRe[kk], bIm[kk], ci);
        if (cplx) {
#pragma unroll
            for (int kk = 0; kk < 4; kk++) cr = WMMA4(aImN[kk], bIm[kk], cr);
#pragma unroll
            for (int kk = 0; kk < 4; kk++) ci = WMMA4(aIm[kk], bRe[kk], ci);
        }

#pragma unroll
        for (int j = 0; j < 8; j++) {          // D: 16x16 f32 C/D layout
            const int M = j + ((lane < 16) ? 0 : 8);
            const int e = base + (M << suf_sh);
            rre[e] = cr[j];
            rim[e] = ci[j];
        }
    }
}

#define STEP_FENCE() do { __threadfence(); __syncthreads(); } while (0)

// ---------------- main simulation: one block per sample ----------------
__global__ __launch_bounds__(256) void simulate(const float* __restrict__ enc,
                                                const float* __restrict__ rot,
                                                const float* __restrict__ cnotT,
                                                float* __restrict__ rho,
                                                float* __restrict__ out) {
    const int smp  = blockIdx.x;
    const int tid  = threadIdx.x;
    const int lane = tid & 31;
    const int wave = tid >> 5;

    float* rre = rho + (size_t)smp * 131072;   // re plane [65536]
    float* rim = rre + 65536;                  // im plane [65536]

    // init rho = |0..0><0..0|
    for (int idx = tid; idx < 131072; idx += 256) rre[idx] = 0.f;
    if (tid == 0) rre[0] = 1.f;
    STEP_FENCE();

    // encode: real pair superops on wires (0,1),(2,3),(4,5),(6,7)
    for (int j = 0; j < 4; j++) {
        const float* T = enc + ((size_t)smp * 4 + j) * 512;
        apply_pair_op(rre, rim, T, T + 256, 2 * j, false, lane, wave);
        STEP_FENCE();
    }

    // layers
    for (int l = 0; l < DEPTH; l++) {
        for (int i = 0; i < NQ - 1; i++) {     // CNOT+2q-noise: constant real superop
            apply_pair_op(rre, rim, cnotT, cnotT + 256, i, false, lane, wave);
            STEP_FENCE();
        }
        for (int j = 0; j < 4; j++) {          // ry*rz*noise pair superops (complex)
            const float* T = rot + ((size_t)l * 4 + j) * 512;
            apply_pair_op(rre, rim, T, T + 256, 2 * j, true, lane, wave);
            STEP_FENCE();
        }
    }

    // expectation <Z_0>: diagonal entries have all m_w in {0,3};
    // addr = 3 * bit-spread(tid); sign from wire-0 bit (tid bit 7).
    __shared__ float red[256];
    int addr = 0;
#pragma unroll
    for (int k = 0; k < 8; k++)
        if ((tid >> k) & 1) addr += 3 << (2 * k);
    float v = rre[addr];
    red[tid] = (tid & 128) ? -v : v;
    __syncthreads();
    for (int off = 128; off > 0; off >>= 1) {
        if (tid < off) red[tid] += red[tid + off];
        __syncthreads();
    }
    if (tid == 0) out[smp] = red[0];
}

// ---------------- launch ----------------
extern "C" void kernel_launch(void* const* d_in, const int* in_sizes, int n_in,
                              void* d_out, int out_size, void* d_ws, size_t ws_size,
                              hipStream_t stream) {
    const float* x = (const float*)d_in[0];   // [32, 8]
    const float* w = (const float*)d_in[1];   // [6, 8, 2]
    float* out = (float*)d_out;               // [32, 1]

    float* ws   = (float*)d_ws;
    float* rho  = ws;                         // 32 * 131072 floats (16 MB, L2-resident)
    float* enc  = ws + (size_t)BATCH * 131072; // 32*4*512
    float* rot  = enc + BATCH * 4 * 512;       // 6*4*512
    float* cnot = rot + DEPTH * 4 * 512;       // 512

    build_superops<<<1, 256, 0, stream>>>(x, w, enc, rot, cnot);
    simulate<<<BATCH, 256, 0, stream>>>(enc, rot, cnot, rho, out);
}